// NMSLayer_38405597561619
// MI455X (gfx1250) — compile-verified
//
#include <hip/hip_runtime.h>
#include <math.h>

// ---------------------------------------------------------------------------
// Problem constants (match reference)
// ---------------------------------------------------------------------------
namespace {
constexpr int   B         = 32;
constexpr int   H         = 128;
constexpr int   W         = 128;
constexpr int   NA        = 9;
constexpr int   C         = NA * 4 + NA + 1;   // 46 channels
constexpr int   N         = H * W;             // 16384 pixels / image
constexpr int   NUM_PROP  = 10;
constexpr float STRIDE    = 16.0f;
constexpr float CLS_TH    = 0.95f;
constexpr float MAX_IOU   = 0.1f;
constexpr float NEGV      = -1000000000.0f;

constexpr int DEC_BLOCK  = 256;                // pixels per decode block
constexpr int DEC_FLOATS = DEC_BLOCK * C;      // 11776 floats = 47104 B LDS
constexpr int DEC_VEC4   = DEC_FLOATS / 4;     // 2944 float4 copies / block

constexpr int NMS_BLOCK  = 1024;               // 32 waves (wave32)
constexpr int PER_THREAD = N / NMS_BLOCK;      // 16 scores per thread
}

// ---------------------------------------------------------------------------
// CDNA5 async global -> LDS copy (gfx1250 ASYNCcnt path)
// Builtin signature (from clang diagnostic): param0 is int-vec4 pointer.
// ---------------------------------------------------------------------------
typedef int vint4 __attribute__((vector_size(4 * sizeof(int))));
typedef __attribute__((address_space(1))) vint4* gvec_ptr_t;
typedef __attribute__((address_space(3))) vint4* lvec_ptr_t;

__device__ __forceinline__ void async_g2l_b128(const float* gsrc, float* ldst) {
  // emits global_load_async_to_lds_b128 (tracked by ASYNCcnt)
  __builtin_amdgcn_global_load_async_to_lds_b128(
      (gvec_ptr_t)gsrc, (lvec_ptr_t)ldst, /*offset=*/0, /*cpol=*/0);
}

__device__ __forceinline__ void wait_async_zero() {
#if __has_builtin(__builtin_amdgcn_s_wait_asynccnt)
  __builtin_amdgcn_s_wait_asynccnt(0);
#else
  asm volatile("s_wait_asynccnt 0" ::: "memory");
#endif
}

// ---------------------------------------------------------------------------
// Kernel 1: decode.  One thread per pixel; each block stages its contiguous
// 47 KB input slab into LDS with async b128 copies (coalesced), then decodes
// from LDS (the per-record access pattern would otherwise be uncoalesced).
// ---------------------------------------------------------------------------
__global__ __launch_bounds__(DEC_BLOCK)
void decode_kernel(const float* __restrict__ in,        // [B*H*W, 46]
                   const float* __restrict__ anchors,   // [9, 2] (ratio, size)
                   float*  __restrict__ scores,         // [B*N]
                   float4* __restrict__ coords,         // [B*N] (y1,x1,y2,x2)
                   float4* __restrict__ boxes)          // [B*N] (b0,b1,b2,b3)
{
  __shared__ float s_in[DEC_FLOATS];

  const int    tid      = threadIdx.x;
  const size_t blockPix = (size_t)blockIdx.x * DEC_BLOCK;
  const float* gsrc     = in + blockPix * C;

  // Stage the whole slab: 2944 float4 chunks over 256 threads (12 passes).
#pragma unroll
  for (int k = 0; k < (DEC_VEC4 + DEC_BLOCK - 1) / DEC_BLOCK; ++k) {
    const int idx = tid + k * DEC_BLOCK;
    if (idx < DEC_VEC4) {
      async_g2l_b128(gsrc + (size_t)idx * 4, &s_in[idx * 4]);
    }
  }
  wait_async_zero();
  __syncthreads();

  const size_t gp  = blockPix + tid;
  const int    rem = (int)(gp % (size_t)N);
  const int    y   = rem / W;
  const int    x   = rem - y * W;

  const float* rec = &s_in[tid * C];

  // argmax over 9 anchor logits (channels 36..44), first-max tie-break
  float best = rec[NA * 4];
  int   aidx = 0;
#pragma unroll
  for (int a = 1; a < NA; ++a) {
    const float v = rec[NA * 4 + a];
    if (v > best) { best = v; aidx = a; }
  }

  // cls (channel 45) zeroed on the border, then thresholded
  float cls = rec[C - 1];
  const bool interior = (y >= 1) & (y <= H - 2) & (x >= 1) & (x <= W - 2);
  if (!interior) cls = 0.0f;
  const float score = (cls > CLS_TH) ? cls : NEGV;

  const float d0 = rec[aidx * 4 + 0];
  const float d1 = rec[aidx * 4 + 1];
  const float d2 = rec[aidx * 4 + 2];
  const float d3 = rec[aidx * 4 + 3];

  const float ratio = anchors[aidx * 2 + 0];
  const float asz   = anchors[aidx * 2 + 1];
  const float a2 = asz;
  const float a3 = asz / ratio;
  const float ax = ((float)x + 0.5f) * STRIDE;
  const float ay = ((float)y + 0.5f) * STRIDE;

  const float b0 = d0 * a2 + ax;
  const float b1 = d1 * a3 + ay;
  const float b2 = expf(d2) * a2;
  const float b3 = expf(d3) * a3;

  boxes[gp]  = make_float4(b0, b1, b2, b3);
  coords[gp] = make_float4(b1 - b3 * 0.5f, b0 - b2 * 0.5f,
                           b1 + b3 * 0.5f, b0 + b2 * 0.5f);
  scores[gp] = score;
}

// ---------------------------------------------------------------------------
// Kernel 2: NMS.  One block per image, 32 wave32 waves.  Scores live in
// registers (16 per thread); coords re-read from L2 each round (8 MB total,
// L2-resident).  Argmax = register scan -> shfl_xor wave reduce -> LDS
// cross-wave combine, with jnp.argmax first-index tie-breaking.
// ---------------------------------------------------------------------------
__global__ __launch_bounds__(NMS_BLOCK)
void nms_kernel(const float*  __restrict__ scores_g,
                const float4* __restrict__ coords_g,
                const float4* __restrict__ boxes_g,
                float*        __restrict__ out)        // [B, NUM_PROP, 4]
{
  __shared__ float s_pv[NMS_BLOCK / 32];
  __shared__ int   s_pi[NMS_BLOCK / 32];
  __shared__ float s_maxv;
  __shared__ int   s_j;

  const int tid  = threadIdx.x;
  const int lane = tid & 31;
  const int wave = tid >> 5;
  const int b    = blockIdx.x;

  const float*  sg = scores_g + (size_t)b * N;
  const float4* cg = coords_g + (size_t)b * N;
  const float4* bg = boxes_g  + (size_t)b * N;

  // scores in registers, strided so lane-adjacent threads stay coalesced
  float sc[PER_THREAD];
#pragma unroll
  for (int k = 0; k < PER_THREAD; ++k) sc[k] = sg[tid + k * NMS_BLOCK];

  for (int it = 0; it < NUM_PROP; ++it) {
    // ---- local argmax (ascending index scan => first-max within thread)
    float best = -3.402823466e38f;
    int   bidx = 0x7fffffff;
#pragma unroll
    for (int k = 0; k < PER_THREAD; ++k) {
      const int   i = tid + k * NMS_BLOCK;
      const float v = sc[k];
      if (v > best) { best = v; bidx = i; }
    }
    // ---- wave32 reduce (greater wins; equal -> smaller index wins)
#pragma unroll
    for (int off = 16; off > 0; off >>= 1) {
      const float ov = __shfl_xor(best, off, 32);
      const int   oi = __shfl_xor(bidx, off, 32);
      if (ov > best || (ov == best && oi < bidx)) { best = ov; bidx = oi; }
    }
    if (lane == 0) { s_pv[wave] = best; s_pi[wave] = bidx; }
    __syncthreads();
    // ---- cross-wave combine in wave 0
    if (wave == 0) {
      float v = s_pv[lane];
      int   i = s_pi[lane];
#pragma unroll
      for (int off = 16; off > 0; off >>= 1) {
        const float ov = __shfl_xor(v, off, 32);
        const int   oi = __shfl_xor(i, off, 32);
        if (ov > v || (ov == v && oi < i)) { v = ov; i = oi; }
      }
      if (lane == 0) { s_maxv = v; s_j = i; }
    }
    __syncthreads();

    const int   j    = s_j;
    const float maxv = s_maxv;
    const bool  ok   = maxv > (NEGV * 0.5f);   // NEG / 2

    const float4 sel    = cg[j];               // uniform address, L2 hit
    const float  area_s = fmaxf(sel.z - sel.x, 0.0f) * fmaxf(sel.w - sel.y, 0.0f);

    if (ok) {
#pragma unroll
      for (int k = 0; k < PER_THREAD; ++k) {
        const int    i = tid + k * NMS_BLOCK;
        const float4 c = cg[i];
        const float area  = fmaxf(c.z - c.x, 0.0f) * fmaxf(c.w - c.y, 0.0f);
        const float iy1   = fmaxf(c.x, sel.x);
        const float ix1   = fmaxf(c.y, sel.y);
        const float iy2   = fminf(c.z, sel.z);
        const float ix2   = fminf(c.w, sel.w);
        const float inter = fmaxf(iy2 - iy1, 0.0f) * fmaxf(ix2 - ix1, 0.0f);
        const float iou   = inter / (area + area_s - inter + 1e-10f);
        if (iou > MAX_IOU || i == j) sc[k] = NEGV;
      }
    }

    if (tid == 0) {
      float4 bx = make_float4(0.0f, 0.0f, 0.0f, 0.0f);
      if (ok) bx = bg[j];
      float* o = out + ((size_t)b * NUM_PROP + it) * 4;
      o[0] = bx.x; o[1] = bx.y; o[2] = bx.z; o[3] = bx.w;
    }
    __syncthreads();   // protect s_pv/s_pi/s_j reuse next round
  }
}

// ---------------------------------------------------------------------------
// Launcher
// ---------------------------------------------------------------------------
extern "C" void kernel_launch(void* const* d_in, const int* in_sizes, int n_in,
                              void* d_out, int out_size, void* d_ws, size_t ws_size,
                              hipStream_t stream) {
  (void)in_sizes; (void)n_in; (void)out_size; (void)ws_size;

  const float* in      = (const float*)d_in[0];   // [B,H,W,46] f32
  const float* anchors = (const float*)d_in[1];   // [9,2] f32

  // workspace layout: scores (2 MB) | coords (8 MB) | boxes (8 MB)
  char* ws = (char*)d_ws;
  float*  scores = (float*)ws;
  float4* coords = (float4*)(ws + sizeof(float) * (size_t)B * N);
  float4* boxes  = (float4*)(ws + sizeof(float) * (size_t)B * N
                                + sizeof(float4) * (size_t)B * N);

  decode_kernel<<<(B * N) / DEC_BLOCK, DEC_BLOCK, 0, stream>>>(
      in, anchors, scores, coords, boxes);
  nms_kernel<<<B, NMS_BLOCK, 0, stream>>>(
      scores, coords, boxes, (float*)d_out);
}